// SparseLinear_11596411699831
// MI455X (gfx1250) — compile-verified
//
#include <hip/hip_runtime.h>

typedef __attribute__((ext_vector_type(2))) float v2f;
typedef __attribute__((ext_vector_type(8))) float v8f;

#define IN_F   4096
#define OUT_F  4096
#define BATCH  1024           // fixed by the reference; enables shift-based addressing
#define NCB    (OUT_F / 16)   // 256 column blocks of width 16
#define NTILES 1024           // tiles for the stable counting sort
#define CHUNK  256            // nnz staged to LDS per block iteration (multiple of 8)

// ---------------------------------------------------------------------------
// Kernel 1: transpose x[B,IN] -> xT[IN,B] so x[:, r] becomes a contiguous row.
// ---------------------------------------------------------------------------
__global__ void k_transpose(const float* __restrict__ x, float* __restrict__ xT) {
    __shared__ float tile[32][33];
    const int ci = blockIdx.x * 32;   // feature (column of x)
    const int bi = blockIdx.y * 32;   // batch
    for (int j = threadIdx.y; j < 32; j += 8)
        tile[j][threadIdx.x] = x[(size_t)(bi + j) * IN_F + ci + threadIdx.x];
    __syncthreads();
    for (int j = threadIdx.y; j < 32; j += 8)
        xT[(size_t)(ci + j) * BATCH + bi + threadIdx.x] = tile[threadIdx.x][j];
}

// ---------------------------------------------------------------------------
// Kernel 2: per-tile histogram of nnz column-block (cb = col >> 4), 256 buckets.
// ---------------------------------------------------------------------------
__global__ void k_hist(const int* __restrict__ cols, int nnz, int tile_sz,
                       int* __restrict__ tileHist) {
    __shared__ int h[256];
    const int t = blockIdx.x;
    h[threadIdx.x] = 0;
    __syncthreads();
    const int beg = t * tile_sz;
    const int end = min(beg + tile_sz, nnz);
    for (int i = beg + (int)threadIdx.x; i < end; i += 256)
        atomicAdd(&h[(cols[i] >> 4) & 255], 1);
    __syncthreads();
    tileHist[t * 256 + threadIdx.x] = h[threadIdx.x];
}

// ---------------------------------------------------------------------------
// Kernel 3: single-block scan. Converts tileHist (counts) into absolute,
// stable scatter offsets; emits bucketStart/bucketCount. Deterministic.
// ---------------------------------------------------------------------------
__global__ void k_scan(int* __restrict__ tileHist, int* __restrict__ bucketStart,
                       int* __restrict__ bucketCount) {
    __shared__ int sc[256];
    const int cb = threadIdx.x;
    int run = 0;
    for (int t = 0; t < NTILES; ++t) {          // serial scan over tiles, per bucket
        int v = tileHist[t * 256 + cb];
        tileHist[t * 256 + cb] = run;
        run += v;
    }
    sc[cb] = run;
    __syncthreads();
    for (int d = 1; d < 256; d <<= 1) {         // Hillis-Steele inclusive scan
        int v = (cb >= d) ? sc[cb - d] : 0;
        __syncthreads();
        sc[cb] += v;
        __syncthreads();
    }
    const int base = sc[cb] - run;              // exclusive prefix
    bucketStart[cb] = base;
    bucketCount[cb] = run;
    __syncthreads();
    for (int t = 0; t < NTILES; ++t)
        tileHist[t * 256 + cb] += base;         // absolute offsets per (tile, bucket)
}

// ---------------------------------------------------------------------------
// Kernel 4: stable scatter into bucket order, packing (row, col&15) into a
// 16-bit code: rc = r | (cl << 12).  r < 4096 fits in 12 bits.
// ---------------------------------------------------------------------------
__global__ void k_scatter(const int* __restrict__ rows, const int* __restrict__ cols,
                          const float* __restrict__ vals, int nnz, int tile_sz,
                          int* __restrict__ tileOff, unsigned short* __restrict__ rc_s,
                          float* __restrict__ vals_s) {
    const int t = blockIdx.x * blockDim.x + threadIdx.x;
    if (t >= NTILES) return;
    const int beg = t * tile_sz;
    const int end = min(beg + tile_sz, nnz);
    for (int i = beg; i < end; ++i) {
        const int c  = cols[i];
        const int cb = (c >> 4) & 255;
        const int p  = tileOff[t * 256 + cb]++;
        rc_s[p]   = (unsigned short)((rows[i] & 0xFFF) | ((c & 15) << 12));
        vals_s[p] = vals[i];
    }
}

// ---------------------------------------------------------------------------
// Kernel 5: WMMA SpMM. One wave owns one 16(batch) x 16(col) output tile.
// D = A(16x4: gathered xT rows) x B(4x16: one-hot * value) + C, two
// independent accumulators (2 WMMA / 8 nnz per iteration) to cover the
// f32 WMMA->WMMA hazard window. No atomics; bias folded into the store.
// ---------------------------------------------------------------------------
__global__ void __launch_bounds__(256)
k_spmm_wmma(const float* __restrict__ xT,
            const int*   __restrict__ bucketStart,
            const int*   __restrict__ bucketCount,
            const unsigned short* __restrict__ rc_s,
            const float* __restrict__ vals_s,
            const float* __restrict__ bias,
            float* __restrict__ out) {
    __shared__ unsigned short lrc[CHUNK];
    __shared__ float          lv[CHUNK];

    const int cb    = blockIdx.x;          // column block (uniform per block)
    const int cbase = cb * 16;
    const int tid   = threadIdx.x;
    const int wave  = tid >> 5;
    const int lane  = tid & 31;
    const int half  = lane >> 4;           // 0: lanes 0-15, 1: lanes 16-31
    const int ln    = lane & 15;
    const int b0    = (blockIdx.y * 8 + wave) * 16;   // batch tile base, per wave

    const int s = bucketStart[cb];
    const int e = s + bucketCount[cb];

    // Per-lane gather base: xT[r][b0+ln] = xp[r << 10]
    const float* __restrict__ xp = xT + (size_t)b0 + ln;

    v8f acc0 = {};
    v8f acc1 = {};
    for (int base = s; base < e; base += CHUNK) {
        __syncthreads();
        for (int i = tid; i < CHUNK; i += 256) {
            const int idx = base + i;
            if (idx < e) { lrc[i] = rc_s[idx]; lv[i] = vals_s[idx]; }
            else         { lrc[i] = 0;         lv[i] = 0.0f;        }
            // stream the next chunk toward L2/L0 (speculative, drop-safe at end)
            __builtin_prefetch(&rc_s[idx + CHUNK], 0, 0);
            __builtin_prefetch(&vals_s[idx + CHUNK], 0, 0);
        }
        __syncthreads();
        const int n    = min(CHUNK, e - base);
        const int kmax = (n + 7) & ~7;      // padded region of LDS is zeroed
        for (int k = 0; k < kmax; k += 8) {
            // ISA A layout (16x4 f32): lanes 0-15 hold K=0,1; lanes 16-31 hold K=2,3
            const int kk0 = k + (half << 1);
            const int kk1 = kk0 + 4;

            // --- WMMA #0: nnz [k, k+4) ---
            const unsigned int rcA = *(const unsigned int*)&lrc[kk0];  // entries kk0, kk0+1
            const int   rA0 = rcA & 0xFFF,          rA1 = (rcA >> 16) & 0xFFF;
            const int   cA0 = (rcA >> 12) & 0xF,    cA1 = rcA >> 28;
            const float vA0 = lv[kk0],              vA1 = lv[kk0 + 1];
            v2f av, bv;
            av.x = xp[rA0 << 10];
            av.y = xp[rA1 << 10];
            bv.x = (cA0 == ln) ? vA0 : 0.0f;
            bv.y = (cA1 == ln) ? vA1 : 0.0f;
            acc0 = __builtin_amdgcn_wmma_f32_16x16x4_f32(
                false, av, false, bv, (short)0, acc0, false, false);

            // --- WMMA #1: nnz [k+4, k+8) (independent accumulator) ---
            const unsigned int rcB = *(const unsigned int*)&lrc[kk1];  // entries kk1, kk1+1
            const int   rB0 = rcB & 0xFFF,          rB1 = (rcB >> 16) & 0xFFF;
            const int   cB0 = (rcB >> 12) & 0xF,    cB1 = rcB >> 28;
            const float vB0 = lv[kk1],              vB1 = lv[kk1 + 1];
            v2f aw, bw;
            aw.x = xp[rB0 << 10];
            aw.y = xp[rB1 << 10];
            bw.x = (cB0 == ln) ? vB0 : 0.0f;
            bw.y = (cB1 == ln) ? vB1 : 0.0f;
            acc1 = __builtin_amdgcn_wmma_f32_16x16x4_f32(
                false, aw, false, bw, (short)0, acc1, false, false);
        }
    }

    // C/D layout: VGPR j -> M=j (lanes 0-15) / M=j+8 (lanes 16-31), N=ln
    const float bz = bias[cbase + ln];
    #pragma unroll
    for (int j = 0; j < 8; ++j) {
        const int m = j + 8 * half;
        out[(size_t)(b0 + m) * OUT_F + cbase + ln] = acc0[j] + acc1[j] + bz;
    }
}

// ---------------------------------------------------------------------------
extern "C" void kernel_launch(void* const* d_in, const int* in_sizes, int n_in,
                              void* d_out, int out_size, void* d_ws, size_t ws_size,
                              hipStream_t stream) {
    (void)n_in; (void)out_size; (void)ws_size;
    const float* x      = (const float*)d_in[0];
    const float* values = (const float*)d_in[1];
    const float* bias   = (const float*)d_in[2];
    const int*   rows   = (const int*)d_in[3];
    const int*   cols   = (const int*)d_in[4];
    const int    nnz    = in_sizes[1];

    // Workspace layout (~18 MB): xT | tileHist/tileOff | bucketStart | bucketCount | sorted COO
    char*  ws  = (char*)d_ws;
    size_t off = 0;
    float* xT          = (float*)(ws + off); off += (size_t)IN_F * BATCH * sizeof(float);
    int*   tileHist    = (int*)  (ws + off); off += (size_t)NTILES * 256 * sizeof(int);
    int*   bucketStart = (int*)  (ws + off); off += 256 * sizeof(int);
    int*   bucketCount = (int*)  (ws + off); off += 256 * sizeof(int);
    unsigned short* rc_s = (unsigned short*)(ws + off);
    off += (((size_t)nnz * sizeof(unsigned short)) + 15) & ~(size_t)15;
    float* vals_s      = (float*)(ws + off); off += (size_t)nnz * sizeof(float);

    const int tile_sz = (nnz + NTILES - 1) / NTILES;

    k_transpose<<<dim3(IN_F / 32, BATCH / 32), dim3(32, 8), 0, stream>>>(x, xT);
    k_hist    <<<NTILES, 256, 0, stream>>>(cols, nnz, tile_sz, tileHist);
    k_scan    <<<1, 256, 0, stream>>>(tileHist, bucketStart, bucketCount);
    k_scatter <<<(NTILES + 255) / 256, 256, 0, stream>>>(rows, cols, values, nnz, tile_sz,
                                                         tileHist, rc_s, vals_s);
    k_spmm_wmma<<<dim3(NCB, BATCH / 128), 256, 0, stream>>>(xT, bucketStart, bucketCount,
                                                            rc_s, vals_s, bias,
                                                            (float*)d_out);
}